// MxCorrelation_18906446037505
// MI455X (gfx1250) — compile-verified
//
#include <hip/hip_runtime.h>

typedef __attribute__((ext_vector_type(2))) float v2f;
typedef __attribute__((ext_vector_type(8))) float v8f;

namespace {
constexpr int Bb = 8, Cc = 128, Hh = 128, Ww = 128;
constexpr int HW_ = Hh * Ww;
constexpr int NGR = 4, GW9 = 9, ND = 81;   // 81 displacements
constexpr int WAVES = 4;
}

// Unconditional load from a clamped (always-valid) address; zero-pad via select
// on the loaded value so no exec-mask branching is generated around the load.
__device__ __forceinline__ v2f load_frag(const float* p, bool pred) {
    const float lo = p[0];      // always in-bounds (clamped address)
    const float hi = p[HW_];
    v2f f;
    f.x = pred ? lo : 0.0f;     // v_cndmask, not a branch
    f.y = pred ? hi : 0.0f;
    return f;
}

__global__ __launch_bounds__(WAVES * 32)
void corr_wmma_kernel(const float* __restrict__ a, const float* __restrict__ b,
                      float* __restrict__ out)
{
    // per-wave scratch for diagonal extraction of the two 16x16 Gram tiles
    __shared__ float gbuf[WAVES][2][16 * 16];

    const int lane = threadIdx.x & 31;
    const int wave = threadIdx.x >> 5;
    const int strip = blockIdx.x * WAVES + wave;     // (b, y, x-tile)

    const int xt  = strip & 7;                       // 8 x-tiles of 16
    const int y   = (strip >> 3) & (Hh - 1);
    const int bat = strip >> 10;                     // 1024 strips per batch
    const int x0  = xt * 16;

    const int half = lane >> 4;                      // 0 or 1
    const int nidx = lane & 15;                      // A row (M) / B,D column (N)

    // ---- preload all A fragments: A[m=i][k=c], lane holds c = 4t + 2*half + j
    const float* abase = a + ((size_t)(bat * Cc) * Hh + y) * Ww + (x0 + nidx);
    v2f afrag[32];
#pragma unroll
    for (int t = 0; t < 32; ++t) {
        const float* ap = abase + (size_t)(4 * t + 2 * half) * HW_;
        afrag[t].x = ap[0];
        afrag[t].y = ap[HW_];
    }

    // N-tile 0 covers Gram columns j=0..15 (x = x0-4+n), tile 1 covers j=8..23.
    const int xb0 = x0 - NGR + nidx;
    const int xb1 = xb0 + 8;
    const bool inx0 = (xb0 >= 0) && (xb0 < Ww);
    const bool inx1 = (xb1 >= 0) && (xb1 < Ww);
    const int xc0 = min(max(xb0, 0), Ww - 1);        // clamped (safe) addresses
    const int xc1 = min(max(xb1, 0), Ww - 1);

    const float* bbat = b + (size_t)(bat * Cc) * Hh * Ww + (size_t)(2 * half) * HW_;

    for (int pi = 0; pi < GW9; ++pi) {               // row displacement p = pi-4
        const int  yb  = y + pi - NGR;
        const bool iny = (yb >= 0) && (yb < Hh);
        const int  yc  = min(max(yb, 0), Hh - 1);    // clamped row

        const float* bp0 = bbat + (size_t)yc * Ww + xc0;
        const float* bp1 = bbat + (size_t)yc * Ww + xc1;
        const bool p0 = iny && inx0;
        const bool p1 = iny && inx1;

        v8f acc0 = {};
        v8f acc1 = {};
#pragma unroll
        for (int t = 0; t < 32; ++t) {               // K loop over C=128, 4 at a time
            const size_t coff = (size_t)(4 * t) * HW_;
            v2f bf0 = load_frag(bp0 + coff, p0);
            v2f bf1 = load_frag(bp1 + coff, p1);
            acc0 = __builtin_amdgcn_wmma_f32_16x16x4_f32(
                       false, afrag[t], false, bf0, (short)0, acc0, false, false);
            acc1 = __builtin_amdgcn_wmma_f32_16x16x4_f32(
                       false, afrag[t], false, bf1, (short)0, acc1, false, false);
        }

        // Spill D tiles to LDS: D[m = r + 8*half][n = nidx]
#pragma unroll
        for (int r = 0; r < 8; ++r) {
            gbuf[wave][0][(r + 8 * half) * 16 + nidx] = acc0[r];
            gbuf[wave][1][(r + 8 * half) * 16 + nidx] = acc1[r];
        }
        __syncthreads();

        // out[i, q] = G[i, i+q] / 128,  q = 0..8 (dq = q-4), j = i+q in [0,23]
#pragma unroll
        for (int t5 = 0; t5 < 5; ++t5) {
            const int k = t5 * 32 + lane;            // 144 outputs per p
            if (k < 144) {
                const int q = k >> 4;
                const int i = k & 15;
                const int j = i + q;
                const float g = (j < 16) ? gbuf[wave][0][i * 16 + j]
                                         : gbuf[wave][1][i * 16 + (j - 8)];
                const int d = pi * GW9 + q;
                out[(((size_t)bat * ND + d) * Hh + y) * Ww + (x0 + i)] =
                    g * (1.0f / 128.0f);
            }
        }
        __syncthreads();                             // protect LDS before next p
    }
}

extern "C" void kernel_launch(void* const* d_in, const int* /*in_sizes*/, int /*n_in*/,
                              void* d_out, int /*out_size*/, void* /*d_ws*/,
                              size_t /*ws_size*/, hipStream_t stream) {
    const float* a = (const float*)d_in[0];
    const float* b = (const float*)d_in[1];
    float* out = (float*)d_out;
    const int strips = Bb * Hh * (Ww / 16);          // 8192 strips of 16 pixels
    dim3 grid(strips / WAVES), block(WAVES * 32);
    corr_wmma_kernel<<<grid, block, 0, stream>>>(a, b, out);
}